// DIMPA_80900003988159
// MI455X (gfx1250) — compile-verified
//
#include <hip/hip_runtime.h>

#define D 64
#define HOP 3

typedef __attribute__((ext_vector_type(2))) float v2f;
typedef __attribute__((ext_vector_type(8))) float v8f;

// ---------------------------------------------------------------------------
// zero a float buffer (128-bit stores)
// ---------------------------------------------------------------------------
__global__ void dimpa_zero(float4* __restrict__ p, int n4) {
  int i = blockIdx.x * blockDim.x + threadIdx.x;
  if (i < n4) p[i] = make_float4(0.f, 0.f, 0.f, 0.f);
}

// ---------------------------------------------------------------------------
// COO scatter SpMM: y[rows[e], :] += vals[e] * x[cols[e], :]
// 16 lanes per edge, float4 per lane -> one half-wave covers all 64 columns.
// Entire working set is L2-resident on MI455X (192 MB), so the gather and the
// native f32 atomic scatter both run at L2 bandwidth.
// ---------------------------------------------------------------------------
__global__ void dimpa_spmm_scatter(const int* __restrict__ rows,
                                   const int* __restrict__ cols,
                                   const float* __restrict__ vals,
                                   const float* __restrict__ x,
                                   float* __restrict__ y, int nedges) {
  int tid = blockIdx.x * blockDim.x + threadIdx.x;
  int e = tid >> 4;  // edge index
  int t = tid & 15;  // column group (4 floats each)
  if (e >= nedges) return;
  // stream edge arrays ahead (global_prefetch_b8, speculative)
  __builtin_prefetch(rows + e + 2048, 0, 0);
  __builtin_prefetch(cols + e + 2048, 0, 0);
  __builtin_prefetch(vals + e + 2048, 0, 0);
  int r = rows[e];
  int c = cols[e];
  float v = vals[e];
  const float4* xr = (const float4*)(x + (size_t)c * D);
  float4 g = xr[t];
  float* yr = y + (size_t)r * D + t * 4;
  unsafeAtomicAdd(yr + 0, v * g.x);
  unsafeAtomicAdd(yr + 1, v * g.y);
  unsafeAtomicAdd(yr + 2, v * g.z);
  unsafeAtomicAdd(yr + 3, v * g.w);
}

// ---------------------------------------------------------------------------
// WMMA combine: dst[:, coff + 0:64] (+)= w[widx] * src[:, 0:64]
// One wave per 16x16 tile. D_tile = sum_{kk=0..3} A_kk x B_kk + C with
//   A_kk = src tile columns [4kk, 4kk+4)          (16x4, f32)
//   B_kk = w * (rows kk of identity, 4x16)        -> exact elementwise scale
// Uses V_WMMA_F32_16X16X4_F32 VGPR layouts:
//   A: lane m in 0..15 (both halves), lanes<16 hold K=0,1; lanes>=16 K=2,3
//   B/C/D: column = lane&15; C/D row = vgpr + 8*(lane>=16)
// ---------------------------------------------------------------------------
__global__ void dimpa_combine_wmma(float* __restrict__ dst, int ld, int coff,
                                   const float* __restrict__ src,
                                   const float* __restrict__ w, int widx,
                                   int ntiles, int accumulate) {
  int wid = (blockIdx.x * blockDim.x + threadIdx.x) >> 5;
  int lane = threadIdx.x & 31;
  if (wid >= ntiles) return;           // whole-wave uniform exit (EXEC stays all-1s)
  int tm = wid >> 2;                   // 16-row tile
  int tc = wid & 3;                    // 16-col tile of the 64 src cols
  float ww = w[widx];                  // scalar (uniform) load of hop weight
  int m = lane & 15;
  int half = lane >> 4;
  int r0 = 2 * half;                   // K base row for this half-wave

  const float* srow = src + (size_t)(tm * 16 + m) * D + tc * 16;
  float* dcol = dst + (size_t)(tm * 16) * ld + coff + tc * 16 + m;

  v8f c;
  if (accumulate) {
#pragma unroll
    for (int v = 0; v < 8; ++v) c[v] = dcol[(size_t)(v + 8 * half) * ld];
  } else {
#pragma unroll
    for (int v = 0; v < 8; ++v) c[v] = 0.0f;
  }

#pragma unroll
  for (int kk = 0; kk < 4; ++kk) {
    v2f a, b;
    a.x = srow[4 * kk + r0 + 0];
    a.y = srow[4 * kk + r0 + 1];
    b.x = (m == 4 * kk + r0 + 0) ? ww : 0.0f;
    b.y = (m == 4 * kk + r0 + 1) ? ww : 0.0f;
    c = __builtin_amdgcn_wmma_f32_16x16x4_f32(false, a, false, b, (short)0, c,
                                              false, false);
  }

#pragma unroll
  for (int v = 0; v < 8; ++v) dcol[(size_t)(v + 8 * half) * ld] = c[v];
}

// ---------------------------------------------------------------------------
// launch
// ---------------------------------------------------------------------------
extern "C" void kernel_launch(void* const* d_in, const int* in_sizes, int n_in,
                              void* d_out, int out_size, void* d_ws, size_t ws_size,
                              hipStream_t stream) {
  const float* x_s    = (const float*)d_in[0];
  const float* x_t    = (const float*)d_in[1];
  const int*   A_rows = (const int*)d_in[2];
  const int*   A_cols = (const int*)d_in[3];
  const float* A_vals = (const float*)d_in[4];
  const int*   At_rows = (const int*)d_in[5];
  const int*   At_cols = (const int*)d_in[6];
  const float* At_vals = (const float*)d_in[7];
  const float* w_s    = (const float*)d_in[8];
  const float* w_t    = (const float*)d_in[9];

  const int N = in_sizes[0] / D;   // 100000 (multiple of 16)
  const int E = in_sizes[2];       // 1200000
  float* out = (float*)d_out;
  const int ld = 2 * D;

  float* bufA = (float*)d_ws;
  float* bufB = bufA + (size_t)N * D;   // ping-pong hop buffers (51.2 MB total)

  const int n4 = (N * D) / 4;
  dim3 zb(256), zg((n4 + 255) / 256);
  dim3 sb(256), sg((E * 16 + 255) / 256);
  const int ntiles = (N / 16) * 4;
  dim3 cb(256), cg((ntiles + 7) / 8);   // 8 waves (tiles) per block

  for (int side = 0; side < 2; ++side) {
    const float* x    = side ? x_t : x_s;
    const int*   rows = side ? At_rows : A_rows;
    const int*   cols = side ? At_cols : A_cols;
    const float* vals = side ? At_vals : A_vals;
    const float* w    = side ? w_t : w_s;
    const int coff = side ? D : 0;

    // feat = w[0] * x   (beta = 0)
    dimpa_combine_wmma<<<cg, cb, 0, stream>>>(out, ld, coff, x, w, 0, ntiles, 0);

    const float* cur = x;
    for (int h = 1; h <= HOP; ++h) {
      float* nxt = (h & 1) ? bufA : bufB;
      dimpa_zero<<<zg, zb, 0, stream>>>((float4*)nxt, n4);
      dimpa_spmm_scatter<<<sg, sb, 0, stream>>>(rows, cols, vals, cur, nxt, E);
      // feat += w[h] * curr
      dimpa_combine_wmma<<<cg, cb, 0, stream>>>(out, ld, coff, nxt, w, h, ntiles, 1);
      cur = nxt;
    }
  }
}